// FutureProposalHead_13486197309985
// MI455X (gfx1250) — compile-verified
//
#include <hip/hip_runtime.h>
#include <hip/hip_bf16.h>
#include <math.h>

typedef __attribute__((ext_vector_type(2))) float v2f;
typedef __attribute__((ext_vector_type(8))) float v8f;

#define D_MODEL   2048
#define SEQ_LEN   8192
#define HIDDEN    64
#define START_IDX 100
#define END_IDX   163
#define SPAN_LEN  64
#define STARTP    164            // END_IDX+1
#define STOPP     4164           // min(8192, 164+4000)
#define NW        15684          // total windows over L in {32,64,96,128}
#define NTILES    ((NW + 15) / 16)   // 981
#define NWPAD     (NTILES * 16)      // 15696 (12 zero pad rows)
#define EPSF      1e-8f
#define PRESSURE  0.7f
#define VIAB_GAP  0.6f
#define DESC_GAP  0.7f
#define REPAIR    0.66f          // 0.6*0.7 + 0.4*0.6
#define TOPK      8
#define CHUNK_ROWS 128
#define NCHUNKS    (SEQ_LEN / CHUNK_ROWS)   // 64
#define FUSK       (4 * D_MODEL)            // 8192

// segment boundaries for flattened (L, off) enumeration
// L=32: [0,3969)  L=64: [3969,7906)  L=96: [7906,11811)  L=128: [11811,15684)
__device__ __forceinline__ void decode_win(int m, int& L, int& off) {
    if (m < 3969)        { L = 32;  off = STARTP + m; }
    else if (m < 7906)   { L = 64;  off = STARTP + (m - 3969); }
    else if (m < 11811)  { L = 96;  off = STARTP + (m - 7906); }
    else                 { L = 128; off = STARTP + (m - 11811); }
}

__device__ __forceinline__ float wred32(float v) {
    v += __shfl_xor(v, 16, 32);
    v += __shfl_xor(v, 8, 32);
    v += __shfl_xor(v, 4, 32);
    v += __shfl_xor(v, 2, 32);
    v += __shfl_xor(v, 1, 32);
    return v;
}
__device__ __forceinline__ float red16(float v) {   // reduce within each 16-lane half
    v += __shfl_xor(v, 8, 32);
    v += __shfl_xor(v, 4, 32);
    v += __shfl_xor(v, 2, 32);
    v += __shfl_xor(v, 1, 32);
    return v;
}
__device__ __forceinline__ float gelu_exact(float x) {
    return 0.5f * x * (1.0f + erff(x * 0.70710678118654752f));
}

// ---------------- K0: per-row inverse norm -----------------------------------
__global__ __launch_bounds__(256) void k_rownorm(const float* __restrict__ h,
                                                 float* __restrict__ invn) {
    __shared__ float red[256];
    int row = blockIdx.x;
    const float* r = h + (size_t)row * D_MODEL;
    float s = 0.f;
    for (int c = threadIdx.x; c < D_MODEL; c += 256) { float v = r[c]; s += v * v; }
    red[threadIdx.x] = s; __syncthreads();
    for (int st = 128; st > 0; st >>= 1) {
        if (threadIdx.x < st) red[threadIdx.x] += red[threadIdx.x + st];
        __syncthreads();
    }
    if (threadIdx.x == 0) invn[row] = 1.0f / fmaxf(sqrtf(red[0]), EPSF);
}

// ---------------- K1a: per-chunk column partial sums --------------------------
__global__ __launch_bounds__(256) void k_chunksum(const float* __restrict__ h,
                                                  const float* __restrict__ invn,
                                                  float* __restrict__ psumH,
                                                  float* __restrict__ psumG) {
    int chunk = blockIdx.x >> 3;
    int c = (blockIdx.x & 7) * 256 + threadIdx.x;
    int r0 = chunk * CHUNK_ROWS;
    float sh = 0.f, sg = 0.f;
    for (int r = r0; r < r0 + CHUNK_ROWS; ++r) {
        float v = h[(size_t)r * D_MODEL + c];
        sh += v;
        sg += v * invn[r];
    }
    psumH[(size_t)chunk * D_MODEL + c] = sh;
    psumG[(size_t)chunk * D_MODEL + c] = sg;
}

// ---------------- K1b: exclusive scan over chunks (per column) ----------------
__global__ __launch_bounds__(256) void k_chunkscan(float* __restrict__ psumH,
                                                   float* __restrict__ psumG) {
    int c = blockIdx.x * 256 + threadIdx.x;
    float aH = 0.f, aG = 0.f;
    for (int k = 0; k < NCHUNKS; ++k) {
        size_t o = (size_t)k * D_MODEL + c;
        float tH = psumH[o], tG = psumG[o];
        psumH[o] = aH; psumG[o] = aG;
        aH += tH; aG += tG;
    }
}

// ---------------- K1c: write csh/csg with chunk prefixes ----------------------
__global__ __launch_bounds__(256) void k_scanwrite(const float* __restrict__ h,
                                                   const float* __restrict__ invn,
                                                   const float* __restrict__ psumH,
                                                   const float* __restrict__ psumG,
                                                   float* __restrict__ csh,
                                                   float* __restrict__ csg) {
    int chunk = blockIdx.x >> 3;
    int c = (blockIdx.x & 7) * 256 + threadIdx.x;
    int r0 = chunk * CHUNK_ROWS;
    float sh = psumH[(size_t)chunk * D_MODEL + c];
    float sg = psumG[(size_t)chunk * D_MODEL + c];
    if (chunk == 0) { csh[c] = 0.f; csg[c] = 0.f; }
    for (int r = r0; r < r0 + CHUNK_ROWS; ++r) {
        float v = h[(size_t)r * D_MODEL + c];
        sh += v;
        sg += v * invn[r];
        size_t o = (size_t)(r + 1) * D_MODEL + c;
        csh[o] = sh;
        csg[o] = sg;
    }
}

// ---------------- K2: first_occ, n_unique, csm, padded w1, feats pad ---------
__global__ __launch_bounds__(64) void k_prep(const int* __restrict__ ids,
                                             const float* __restrict__ w1,
                                             float* __restrict__ scal,   // [0]=n_unique
                                             float* __restrict__ focc,
                                             int* __restrict__ csm,
                                             float* __restrict__ w1pad,  // 12 x 64
                                             float* __restrict__ feats) {
    int a = threadIdx.x;             // 64 anchors
    int aid = ids[START_IDX + a];
    int first = 0;
    for (int j = 0; j < SPAN_LEN; ++j) {
        if (ids[START_IDX + j] == aid) { first = j; break; }
    }
    focc[a] = (first == a) ? 1.0f : 0.0f;
    int* row = csm + (size_t)a * (SEQ_LEN + 1);
    int cnt = 0;
    row[0] = 0;
    for (int t = 0; t < SEQ_LEN; ++t) {
        cnt += (ids[t] == aid) ? 1 : 0;
        row[t + 1] = cnt;
    }
    // padded copy of w1: rows 10,11 zero -> unconditional WMMA B loads
    for (int idx = a; idx < 12 * HIDDEN; idx += 64) {
        int r = idx / HIDDEN;
        w1pad[idx] = (r < 10) ? w1[idx] : 0.f;
    }
    // zero the 12 pad rows of feats (rows NW..NWPAD-1)
    for (int idx = a; idx < (NWPAD - NW) * 12; idx += 64) {
        feats[(size_t)NW * 12 + idx] = 0.f;
    }
    __syncthreads();
    if (a == 0) {
        float s = 0.f;
        for (int j = 0; j < SPAN_LEN; ++j) s += focc[j];
        scal[0] = fmaxf(s, 1.0f);
    }
}

// ---------------- K3: per-window features (one wave per window) --------------
__global__ __launch_bounds__(256) void k_windows(const float* __restrict__ h,
                                                 const int* __restrict__ ids,
                                                 const float* __restrict__ anchor,
                                                 const float* __restrict__ csh,
                                                 const float* __restrict__ csg,
                                                 const int* __restrict__ csm,
                                                 const float* __restrict__ focc,
                                                 const float* __restrict__ scal,
                                                 float* __restrict__ feats,
                                                 float* __restrict__ heur,
                                                 float* __restrict__ validf,
                                                 int* __restrict__ roots) {
    __shared__ float s_anchor[D_MODEL];
    __shared__ float s_ad[D_MODEL];
    {
        // Stage the block-shared anchor vector into LDS with CDNA5 async copy
        // (GLOBAL_LOAD_ASYNC_TO_LDS_B32, tracked by ASYNCcnt), and build the
        // shared ad_mean vector with plain loads.
        const float* aS = h + (size_t)START_IDX * D_MODEL;
        const float* aE = h + (size_t)END_IDX * D_MODEL;
        for (int c = threadIdx.x; c < D_MODEL; c += 256) {
            unsigned loff = (unsigned)(size_t)&s_anchor[c];
            const float* gp = anchor + c;
            asm volatile("global_load_async_to_lds_b32 %0, %1, off"
                         :: "v"(loff), "v"(gp) : "memory");
            s_ad[c] = (aE[c] - aS[c]) * (1.0f / 63.0f);
        }
        asm volatile("s_wait_asynccnt 0" ::: "memory");
    }
    __syncthreads();

    int lane = threadIdx.x & 31;
    int win = blockIdx.x * 8 + (threadIdx.x >> 5);
    if (win >= NW) return;
    int L, off; decode_win(win, L, off);
    const float invL   = 1.0f / (float)L;
    const float invLm1 = 1.0f / (float)(L - 1);
    const int tail_len = STOPP - (off + L);
    const float invT   = 1.0f / (float)max(tail_len, 1);

    const float* c0 = csh + (size_t)off * D_MODEL;
    const float* c1 = csh + (size_t)(off + L) * D_MODEL;
    const float* cS = csh + (size_t)STOPP * D_MODEL;
    const float* g0 = csg + (size_t)off * D_MODEL;
    const float* g1 = csg + (size_t)(off + L) * D_MODEL;
    const float* h0 = h + (size_t)off * D_MODEL;
    const float* h1 = h + (size_t)(off + L - 1) * D_MODEL;

    float s_aw=0, s_ww=0, s_aa=0, s_dm=0, s_d2=0, s_a2=0, s_gw=0, s_tw=0, s_tt=0;
    for (int c = lane; c < D_MODEL; c += 32) {
        float wm = (c1[c] - c0[c]) * invL;
        float a  = s_anchor[c];
        s_aw += a * wm; s_ww += wm * wm; s_aa += a * a;
        float ad  = s_ad[c];
        float wdm = (h1[c] - h0[c]) * invLm1;
        s_dm += ad * wdm; s_d2 += wdm * wdm; s_a2 += ad * ad;
        float gb = (g1[c] - g0[c]) * invL;
        s_gw += gb * wm;
        float tl = (cS[c] - c1[c]) * invT;
        s_tw += wm * tl; s_tt += tl * tl;
    }
    float aw = wred32(s_aw), ww = wred32(s_ww), aa = wred32(s_aa);
    float dm = wred32(s_dm), d2 = wred32(s_d2), a2 = wred32(s_a2);
    float gw = wred32(s_gw), tw = wred32(s_tw), tt = wred32(s_tt);

    // overlap
    float s_ov = 0.f;
    for (int a = lane; a < SPAN_LEN; a += 32) {
        const int* row = csm + (size_t)a * (SEQ_LEN + 1);
        int cont = (row[off + L] - row[off]) > 0 ? 1 : 0;
        s_ov += focc[a] * (float)cont;
    }
    float ov = wred32(s_ov);

    if (lane == 0) {
        float nuniq = scal[0];
        float nrm_wm = sqrtf(ww);
        float mean_sim = (aw / fmaxf(sqrtf(aa) * nrm_wm, EPSF) + 1.0f) * 0.5f;
        float contrast = 1.0f - mean_sim;
        float trans    = (dm / fmaxf(sqrtf(a2) * sqrtf(d2), EPSF) + 1.0f) * 0.5f;
        float coh      = (gw / fmaxf(nrm_wm, EPSF) + 1.0f) * 0.5f;
        float tail_sup = (tail_len > 0)
            ? (tw / fmaxf(nrm_wm * sqrtf(tt), EPSF) + 1.0f) * 0.5f
            : coh;
        float overlap = ov / nuniq;
        float dd = 1.0f / (1.0f + (float)(off - END_IDX) / (float)SPAN_LEN);

        float* f = feats + (size_t)win * 12;
        f[0] = contrast; f[1] = mean_sim; f[2] = trans; f[3] = coh; f[4] = tail_sup;
        f[5] = overlap;  f[6] = dd;       f[7] = PRESSURE; f[8] = VIAB_GAP; f[9] = DESC_GAP;
        f[10] = 0.f; f[11] = 0.f;

        float conflict = 0.55f * contrast + 0.25f * (1.0f - trans) + 0.2f * (1.0f - overlap);
        float plaus    = 0.45f * coh + 0.35f * tail_sup + 0.2f * dd;
        heur[win] = 2.4f * (conflict - 0.35f) + 2.0f * (plaus - 0.55f)
                  + 1.4f * (REPAIR - 0.5f) + 0.5f * (DESC_GAP - 0.35f);
        validf[win] = (conflict >= 0.18f) ? 1.0f : 0.0f;   // REPAIR>=0.35 always true
        roots[win] = ids[off + L - 1];
    }
}

// ---------------- K4: feats MLP via WMMA f32 16x16x4 -------------------------
__global__ __launch_bounds__(256) void k_featmlp(const float* __restrict__ feats,
                                                 const float* __restrict__ heur,
                                                 const float* __restrict__ validf,
                                                 const float* __restrict__ w1pad,
                                                 const float* __restrict__ b1,
                                                 const float* __restrict__ w2,
                                                 const float* __restrict__ b2,
                                                 float* __restrict__ scores) {
    int lane = threadIdx.x & 31;
    int tile = blockIdx.x * 8 + (threadIdx.x >> 5);
    if (tile >= NTILES) return;
    int mb = tile * 16;
    int ml = lane & 15;
    int kg = (lane >> 4) << 1;
    int mrow = mb + ml;

    v8f c[4];
    for (int nt = 0; nt < 4; ++nt)
        c[nt] = (v8f){0.f,0.f,0.f,0.f,0.f,0.f,0.f,0.f};

    for (int k = 0; k < 12; k += 4) {
        int kk = k + kg;
        v2f a = *(const v2f*)&feats[(size_t)mrow * 12 + kk];   // padded: no guard
        for (int nt = 0; nt < 4; ++nt) {
            int n = nt * 16 + ml;
            v2f b;
            b.x = w1pad[kk * HIDDEN + n];
            b.y = w1pad[(kk + 1) * HIDDEN + n];
            c[nt] = __builtin_amdgcn_wmma_f32_16x16x4_f32(
                false, a, false, b, (short)0, c[nt], false, false);
        }
    }

    float part[8];
    for (int v = 0; v < 8; ++v) part[v] = 0.f;
    for (int nt = 0; nt < 4; ++nt) {
        int n = nt * 16 + ml;
        float b1n = b1[n], w2n = w2[n];
        for (int v = 0; v < 8; ++v)
            part[v] += gelu_exact(c[nt][v] + b1n) * w2n;
    }
    for (int v = 0; v < 8; ++v) part[v] = red16(part[v]);

    if ((lane & 15) == 0) {
        int mo = mb + ((lane >> 4) << 3);  // lane0 -> rows mb..mb+7, lane16 -> mb+8..mb+15
        float b2s = b2[0];
        for (int v = 0; v < 8; ++v) {
            int m = mo + v;
            if (m < NW) {
                float learned = 0.25f * (part[v] + b2s);
                float sc = 1.0f / (1.0f + expf(-(heur[m] + learned)));
                scores[m] = (validf[m] > 0.5f) ? sc : -__builtin_inff();
            }
        }
    }
}

// ---------------- K5: top-k, softmax, gather top_repr, build fusion ----------
__device__ __forceinline__ float fusion_val(int m, int c,
                                            const float* __restrict__ anchor,
                                            const float* __restrict__ tr) {
    if (m >= TOPK) return 0.f;
    int seg = c >> 11, cc = c & (D_MODEL - 1);
    float ar = anchor[cc];
    if (seg == 0) return ar;
    float t = tr[m * D_MODEL + cc];
    if (seg == 1) return t;
    if (seg == 2) return t - ar;
    return t * ar;
}

__global__ __launch_bounds__(256) void k_topk(const float* __restrict__ scores,
                                              const int* __restrict__ roots,
                                              const float* __restrict__ csh,
                                              const float* __restrict__ anchor,
                                              float* __restrict__ wvals,
                                              float* __restrict__ toprep,
                                              float* __restrict__ fusb,
                                              float* __restrict__ out) {
    __shared__ float sv[256];
    __shared__ int   si[256];
    __shared__ float cv[TOPK];
    __shared__ int   ci[TOPK];
    int tid = threadIdx.x;
    for (int p = 0; p < TOPK; ++p) {
        float best = -__builtin_inff();
        int   bi = 0x7fffffff;
        for (int m = tid; m < NW; m += 256) {
            bool skip = false;
            for (int q = 0; q < p; ++q) if (ci[q] == m) skip = true;
            if (skip) continue;
            float v = scores[m];
            if (v > best || (v == best && m < bi)) { best = v; bi = m; }
        }
        sv[tid] = best; si[tid] = bi; __syncthreads();
        for (int st = 128; st > 0; st >>= 1) {
            if (tid < st) {
                if (sv[tid + st] > sv[tid] ||
                    (sv[tid + st] == sv[tid] && si[tid + st] < si[tid])) {
                    sv[tid] = sv[tid + st]; si[tid] = si[tid + st];
                }
            }
            __syncthreads();
        }
        if (tid == 0) { cv[p] = sv[0]; ci[p] = si[0]; }
        __syncthreads();
    }
    if (tid == 0) {
        float mx = cv[0], s = 0.f, e[TOPK];
        for (int i = 0; i < TOPK; ++i) { e[i] = expf(cv[i] - mx); s += e[i]; }
        for (int i = 0; i < TOPK; ++i) wvals[i] = e[i] / s;
        out[D_MODEL]     = cv[0];
        out[D_MODEL + 1] = (float)roots[ci[0]];
    }
    __syncthreads();
    for (int e = tid; e < TOPK * D_MODEL; e += 256) {
        int i = e >> 11, cc = e & (D_MODEL - 1);
        int L, off; decode_win(ci[i], L, off);
        toprep[e] = (csh[(size_t)(off + L) * D_MODEL + cc] -
                     csh[(size_t)off * D_MODEL + cc]) / (float)L;
    }
    __syncthreads();
    // build the 16 x 8192 fusion matrix (rows 8..15 zero) for the WMMA GEMM
    for (int e = tid; e < 16 * FUSK; e += 256) {
        int m = e >> 13;             // e / 8192
        int c = e & (FUSK - 1);
        fusb[e] = fusion_val(m, c, anchor, toprep);
    }
}

// ---------------- K6: fusion MLP via WMMA, weighted residual sum -------------
__global__ __launch_bounds__(128) void k_fuse(const float* __restrict__ fusb,
                                              const float* __restrict__ toprep,
                                              const float* __restrict__ wvals,
                                              const float* __restrict__ wd1,
                                              const float* __restrict__ bd1,
                                              const float* __restrict__ wd2,
                                              const float* __restrict__ bd2,
                                              float* __restrict__ out) {
    __shared__ float hsh[16 * HIDDEN];
    int tid = threadIdx.x, lane = tid & 31, w = tid >> 5;
    int ml = lane & 15;
    int kg = (lane >> 4) << 1;
    int ncol = w * 16 + ml;

    // GEMM1: (16 x 8192) fusion @ (8192 x 64) wd1
    v8f c1 = (v8f){0.f,0.f,0.f,0.f,0.f,0.f,0.f,0.f};
    for (int k = 0; k < FUSK; k += 4) {
        int kk = k + kg;
        __builtin_prefetch(&wd1[(size_t)(kk + 4) * HIDDEN + ncol], 0, 1);
        v2f a = *(const v2f*)&fusb[(size_t)ml * FUSK + kk];
        v2f b;
        b.x = wd1[(size_t)kk * HIDDEN + ncol];
        b.y = wd1[(size_t)(kk + 1) * HIDDEN + ncol];
        c1 = __builtin_amdgcn_wmma_f32_16x16x4_f32(
            false, a, false, b, (short)0, c1, false, false);
    }
    // hidden = gelu(C + bd1) -> LDS (D layout: lane<16 -> M=v, lane>=16 -> M=v+8)
    {
        float bn = bd1[ncol];
        for (int v = 0; v < 8; ++v) {
            int m = (lane < 16) ? v : v + 8;
            hsh[m * HIDDEN + ncol] = gelu_exact(c1[v] + bn);
        }
    }
    __syncthreads();

    float wv[TOPK];
    for (int i = 0; i < TOPK; ++i) wv[i] = wvals[i];

    // GEMM2: (16 x 64) hidden @ (64 x 2048) wd2, 128 N-tiles over 4 waves
    for (int t = w; t < D_MODEL / 16; t += 4) {
        int nt = t * 16;
        v8f c2 = (v8f){0.f,0.f,0.f,0.f,0.f,0.f,0.f,0.f};
        for (int k = 0; k < HIDDEN; k += 4) {
            int kk = k + kg;
            v2f a, b;
            a.x = hsh[ml * HIDDEN + kk];
            a.y = hsh[ml * HIDDEN + kk + 1];
            b.x = wd2[(size_t)kk * D_MODEL + nt + ml];
            b.y = wd2[(size_t)(kk + 1) * D_MODEL + nt + ml];
            c2 = __builtin_amdgcn_wmma_f32_16x16x4_f32(
                false, a, false, b, (short)0, c2, false, false);
        }
        if (lane < 16) {   // rows 0..7 live in lanes 0..15
            int n = nt + ml;
            float bd = bd2[n];
            float p = 0.f;
            for (int v = 0; v < TOPK; ++v) {
                float tr = toprep[v * D_MODEL + n];
                p += wv[v] * (tr + 0.1f * (c2[v] + bd));
            }
            out[n] = p;
        }
    }
}

// ---------------- launch ------------------------------------------------------
extern "C" void kernel_launch(void* const* d_in, const int* in_sizes, int n_in,
                              void* d_out, int out_size, void* d_ws, size_t ws_size,
                              hipStream_t stream) {
    const float* h      = (const float*)d_in[0];
    const int*   ids    = (const int*)d_in[1];
    const float* anchor = (const float*)d_in[2];
    const float* w1 = (const float*)d_in[3];
    const float* b1 = (const float*)d_in[4];
    const float* w2 = (const float*)d_in[5];
    const float* b2 = (const float*)d_in[6];
    const float* wd1 = (const float*)d_in[7];
    const float* bd1 = (const float*)d_in[8];
    const float* wd2 = (const float*)d_in[9];
    const float* bd2 = (const float*)d_in[10];
    float* out = (float*)d_out;

    char* p = (char*)d_ws;
    auto take = [&](size_t bytes) {
        char* r = p;
        p += (bytes + 255) & ~(size_t)255;
        return (void*)r;
    };
    float* invn   = (float*)take(sizeof(float) * SEQ_LEN);
    float* csh    = (float*)take(sizeof(float) * (size_t)(SEQ_LEN + 1) * D_MODEL);
    float* csg    = (float*)take(sizeof(float) * (size_t)(SEQ_LEN + 1) * D_MODEL);
    float* psumH  = (float*)take(sizeof(float) * (size_t)NCHUNKS * D_MODEL);
    float* psumG  = (float*)take(sizeof(float) * (size_t)NCHUNKS * D_MODEL);
    float* scal   = (float*)take(sizeof(float) * 4);
    float* focc   = (float*)take(sizeof(float) * SPAN_LEN);
    int*   csm    = (int*)  take(sizeof(int) * (size_t)SPAN_LEN * (SEQ_LEN + 1));
    float* feats  = (float*)take(sizeof(float) * (size_t)NWPAD * 12);
    float* heur   = (float*)take(sizeof(float) * NW);
    float* validf = (float*)take(sizeof(float) * NW);
    int*   roots  = (int*)  take(sizeof(int) * NW);
    float* scores = (float*)take(sizeof(float) * NW);
    float* wvals  = (float*)take(sizeof(float) * TOPK);
    float* toprep = (float*)take(sizeof(float) * TOPK * D_MODEL);
    float* fusb   = (float*)take(sizeof(float) * 16 * FUSK);
    float* w1pad  = (float*)take(sizeof(float) * 12 * HIDDEN);

    k_rownorm<<<SEQ_LEN, 256, 0, stream>>>(h, invn);
    k_chunksum<<<NCHUNKS * 8, 256, 0, stream>>>(h, invn, psumH, psumG);
    k_chunkscan<<<D_MODEL / 256, 256, 0, stream>>>(psumH, psumG);
    k_scanwrite<<<NCHUNKS * 8, 256, 0, stream>>>(h, invn, psumH, psumG, csh, csg);
    k_prep<<<1, 64, 0, stream>>>(ids, w1, scal, focc, csm, w1pad, feats);
    k_windows<<<(NW + 7) / 8, 256, 0, stream>>>(h, ids, anchor, csh, csg, csm,
                                                focc, scal, feats, heur, validf, roots);
    k_featmlp<<<(NTILES + 7) / 8, 256, 0, stream>>>(feats, heur, validf,
                                                    w1pad, b1, w2, b2, scores);
    k_topk<<<1, 256, 0, stream>>>(scores, roots, csh, anchor, wvals, toprep, fusb, out);
    k_fuse<<<1, 128, 0, stream>>>(fusb, toprep, wvals, wd1, bd1, wd2, bd2, out);
}